// BPSConditionTokenizer_54099408061131
// MI455X (gfx1250) — compile-verified
//
#include <hip/hip_runtime.h>
#include <hip/hip_bf16.h>
#include <math.h>

// BPS condition tokenizer for MI455X (gfx1250, wave32).
// cross[p,n] = basis[p] . pc[n] computed with V_WMMA_F32_16X16X4_F32 (K=3 padded to 4).
// argmin_n (b_sq + pc_sq - 2*cross) == argmin_n (pc_sq - 2*cross)  (b_sq const per row).

typedef __attribute__((ext_vector_type(2))) float v2f;
typedef __attribute__((ext_vector_type(8))) float v8f;

#if defined(__HIP_DEVICE_COMPILE__)
#if !__has_builtin(__builtin_amdgcn_wmma_f32_16x16x4_f32)
#error "gfx1250 device builtin __builtin_amdgcn_wmma_f32_16x16x4_f32 not available"
#endif
#endif

#define BPS_B 16
#define BPS_N 4096
#define BPS_P 4096

__launch_bounds__(256)
__global__ void bps_tokenize_kernel(const float* __restrict__ pc,     // [B, N, 3]
                                    const float* __restrict__ basis,  // [P, 3]
                                    float* __restrict__ out)          // [B, P, 4]
{
    __shared__ float spc[BPS_N * 3];  // 48 KB: this batch's full point cloud

    const int b    = blockIdx.y;
    const int tid  = threadIdx.x;
    const int wave = tid >> 5;
    const int lane = tid & 31;
    const int hf   = lane >> 4;   // which half of the wave (K=0,1 vs K=2,3 operand rows)
    const int m    = lane & 15;   // row/col index within the 16-wide tile

    // Cooperative stage: pc[b] (12288 floats) -> LDS via float4 (3072 vec4 loads).
    {
        const float4* __restrict__ src = (const float4*)(pc + (size_t)b * BPS_N * 3);
        float4* dst = (float4*)spc;
        #pragma unroll
        for (int i = 0; i < (BPS_N * 3 / 4) / 256; ++i)
            dst[tid + i * 256] = src[tid + i * 256];
    }
    __syncthreads();

    // This wave's 16 basis rows.
    const int p_tile = blockIdx.x * 128 + wave * 16;

    // A operand: 16x4 f32 (M x K), K padded with 0.
    // Layout: lanes 0-15 hold {K=0,K=1} = {x,y}; lanes 16-31 hold {K=2,K=3} = {z,0}.
    const float abx = basis[3 * (p_tile + m) + 0];
    const float aby = basis[3 * (p_tile + m) + 1];
    const float abz = basis[3 * (p_tile + m) + 2];
    v2f A;
    A.x = hf ? abz : abx;
    A.y = hf ? 0.0f : aby;

    float best[8];
    int   bidx[8];
    #pragma unroll
    for (int r = 0; r < 8; ++r) { best[r] = __builtin_inff(); bidx[r] = 0; }

    for (int n0 = 0; n0 < BPS_N; n0 += 16) {
        const int n = n0 + m;
        // Every lane reads its column's point (lanes l and l+16 broadcast-read same addr).
        const float x = spc[3 * n + 0];
        const float y = spc[3 * n + 1];
        const float z = spc[3 * n + 2];
        const float pcsq = __builtin_fmaf(x, x, __builtin_fmaf(y, y, z * z));

        // B operand: 4x16 f32 (K x N), mirrors A's K striping across wave halves.
        v2f Bop;
        Bop.x = hf ? z    : x;
        Bop.y = hf ? 0.0f : y;

        v8f c = {};
#if defined(__HIP_DEVICE_COMPILE__)
        c = __builtin_amdgcn_wmma_f32_16x16x4_f32(
                /*neg_a=*/false, A, /*neg_b=*/false, Bop,
                /*c_mod=*/(short)0, c, /*reuse_a=*/false, /*reuse_b=*/false);
#endif

        // C/D layout: reg r, lanes 0-15 -> (M=r, N=lane); lanes 16-31 -> (M=r+8, N=lane-16).
        #pragma unroll
        for (int r = 0; r < 8; ++r) {
            const float key = __builtin_fmaf(-2.0f, c[r], pcsq);  // pc_sq - 2*cross
            const bool lt = key < best[r];
            best[r] = lt ? key : best[r];
            bidx[r] = lt ? n   : bidx[r];
        }
    }

    // Cross-lane argmin over the 16 columns of each half-wave, then emit tokens.
    #pragma unroll
    for (int r = 0; r < 8; ++r) {
        float bv = best[r];
        int   bi = bidx[r];
        #pragma unroll
        for (int off = 1; off < 16; off <<= 1) {
            const float ov = __shfl_xor(bv, off, 32);
            const int   oi = __shfl_xor(bi, off, 32);
            // lexicographic (key, idx) min -> first-occurrence argmin semantics
            const bool take = (ov < bv) || ((ov == bv) && (oi < bi));
            bv = take ? ov : bv;
            bi = take ? oi : bi;
        }
        if (m == r) {  // one lane per half writes row M = r + 8*hf
            const int p = p_tile + hf * 8 + r;
            const float nx = spc[3 * bi + 0];
            const float ny = spc[3 * bi + 1];
            const float nz = spc[3 * bi + 2];
            const float gx = basis[3 * p + 0];
            const float gy = basis[3 * p + 1];
            const float gz = basis[3 * p + 2];
            const float dx = nx - gx;
            const float dy = ny - gy;
            const float dz = nz - gz;
            const float dist = sqrtf(__builtin_fmaf(dx, dx, __builtin_fmaf(dy, dy, dz * dz)));
            ((float4*)out)[(size_t)b * BPS_P + p] = make_float4(dist, dx, dy, dz);
        }
    }
}

extern "C" void kernel_launch(void* const* d_in, const int* in_sizes, int n_in,
                              void* d_out, int out_size, void* d_ws, size_t ws_size,
                              hipStream_t stream) {
    const float* pc    = (const float*)d_in[0];  // [16, 4096, 3] f32
    const float* basis = (const float*)d_in[1];  // [4096, 3] f32
    float*       out   = (float*)d_out;          // [16, 4096, 4] f32

    (void)in_sizes; (void)n_in; (void)out_size; (void)d_ws; (void)ws_size;

    dim3 grid(BPS_P / 128, BPS_B);   // 32 x 16 blocks; 8 waves/block, 16 basis rows/wave
    dim3 block(256);
    bps_tokenize_kernel<<<grid, block, 0, stream>>>(pc, basis, out);
}